// Criterion_50706383897362
// MI455X (gfx1250) — compile-verified
//
#include <hip/hip_runtime.h>

#define NUM_CLASSES 80
#define NB_F 2048      // blocks for focal pass
#define NB_B 256       // blocks for box pass
#define BLK  256       // 8 waves (wave32)

typedef __attribute__((ext_vector_type(2))) float v2f;
typedef __attribute__((ext_vector_type(8))) float v8f;

// Sum v across all 32 lanes of the wave using one V_WMMA_F32_16X16X4_F32.
// A = ones(16x4), B VGPR0 = v (covers 2 full rows of the 4x16 B matrix),
// B VGPR1 = 0  =>  D[m][n] = v_n + v_{n+16} for every m.
// c[0] then holds v_{L%16}+v_{L%16+16} in lane L; 4 xor-shuffles finish.
// Requires EXEC == all ones (all call sites are full 256-thread blocks,
// called unconditionally after loops reconverge).
__device__ __forceinline__ float wave_sum32(float v) {
    v2f a; a[0] = 1.0f; a[1] = 1.0f;
    v2f b; b[0] = v;    b[1] = 0.0f;
    v8f c = {};
    c = __builtin_amdgcn_wmma_f32_16x16x4_f32(false, a, false, b,
                                              (short)0, c, false, false);
    float s = c[0];
    s += __shfl_xor(s, 1, 32);
    s += __shfl_xor(s, 2, 32);
    s += __shfl_xor(s, 4, 32);
    s += __shfl_xor(s, 8, 32);
    return s;   // total of all 32 lanes, in every lane
}

// Deterministic block reduction; result valid on thread 0.
__device__ __forceinline__ float block_sum(float v, float* smem) {
    float w = wave_sum32(v);
    __syncthreads();                       // protect smem across repeated calls
    if ((threadIdx.x & 31) == 0) smem[threadIdx.x >> 5] = w;
    __syncthreads();
    float t = 0.0f;
    if (threadIdx.x == 0) {
        #pragma unroll
        for (int k = 0; k < BLK / 32; ++k) t += smem[k];
    }
    return t;
}

// ---------------- Pass 1: sigmoid focal loss over all (anchor, class) ----------------
__global__ void focal_kernel(const float4* __restrict__ pcls,
                             const int* __restrict__ tcls,
                             const unsigned char* __restrict__ mask,
                             float* __restrict__ partial, int n4) {
    const int tid    = blockIdx.x * blockDim.x + threadIdx.x;
    const int stride = gridDim.x * blockDim.x;
    float acc = 0.0f;

    for (int i = tid; i < n4; i += stride) {
        const int anchor = i / 20;                 // 20 float4 per anchor (80 classes)
        const int cbase  = (i - anchor * 20) * 4;  // first class index of this float4
        const int tc     = tcls[anchor];
        const bool valid = (tc >= 0) && (mask[anchor] != 0);
        if (!valid) continue;                      // reference multiplies focal by valid
        const bool fg = (tc != NUM_CLASSES);       // tc >= 0 already known

        const float4 x4 = pcls[i];
        float xs[4] = {x4.x, x4.y, x4.z, x4.w};
        float s = 0.0f;
        #pragma unroll
        for (int j = 0; j < 4; ++j) {
            const float x  = xs[j];
            const float e  = __expf(-fabsf(x));        // v_exp_f32
            const float r  = 1.0f / (1.0f + e);        // rcp + NR
            const float p  = (x >= 0.0f) ? r : e * r;  // sigmoid(x), stable
            const float ce0 = fmaxf(x, 0.0f) - __logf(r); // = max(x,0)+log1p(e)
            float l;
            if (fg && (cbase + j) == tc) {             // the single one-hot class
                const float q = 1.0f - p;
                l = 0.25f * (ce0 - x) * q * q;         // alpha * ce(t=1) * (1-p)^2
            } else {
                l = 0.75f * ce0 * p * p;               // (1-alpha) * ce(t=0) * p^2
            }
            s += l;
        }
        acc += s;
    }

    __shared__ float smem[BLK / 32];
    const float t = block_sum(acc, smem);
    if (threadIdx.x == 0) partial[blockIdx.x] = t;
}

// ---------------- Pass 2: paired GIoU + foreground count ----------------
__global__ void box_kernel(const float4* __restrict__ pbox,
                           const float4* __restrict__ tbox,
                           const int* __restrict__ tcls,
                           float* __restrict__ pgiou,
                           float* __restrict__ pcnt, int n) {
    const int tid    = blockIdx.x * blockDim.x + threadIdx.x;
    const int stride = gridDim.x * blockDim.x;
    float ag = 0.0f, ac = 0.0f;

    for (int i = tid; i < n; i += stride) {
        const int tc = tcls[i];
        const bool fg = (tc >= 0) && (tc != NUM_CLASSES);
        if (!fg) continue;
        const float4 b1 = pbox[i];
        const float4 b2 = tbox[i];

        const float iw = fmaxf(fminf(b1.z, b2.z) - fmaxf(b1.x, b2.x), 0.0f);
        const float ih = fmaxf(fminf(b1.w, b2.w) - fmaxf(b1.y, b2.y), 0.0f);
        const float inter = iw * ih;
        const float a1 = (b1.z - b1.x) * (b1.w - b1.y);
        const float a2 = (b2.z - b2.x) * (b2.w - b2.y);
        const float uni = a1 + a2 - inter;
        const float iou = inter / uni;
        const float cw = fmaxf(b1.z, b2.z) - fminf(b1.x, b2.x);
        const float ch = fmaxf(b1.w, b2.w) - fminf(b1.y, b2.y);
        const float areac = cw * ch;
        const float giou = iou - (areac - uni) / areac;

        ag += 1.0f - giou;
        ac += 1.0f;
    }

    __shared__ float smem[BLK / 32];
    const float tg = block_sum(ag, smem);
    const float tc_ = block_sum(ac, smem);
    if (threadIdx.x == 0) {
        pgiou[blockIdx.x] = tg;
        pcnt[blockIdx.x]  = tc_;
    }
}

// ---------------- Pass 3: deterministic finalize ----------------
__global__ void finalize_kernel(const float* __restrict__ pf,
                                const float* __restrict__ pg,
                                const float* __restrict__ pc,
                                float* __restrict__ out) {
    float af = 0.0f, ag = 0.0f, ac = 0.0f;
    for (int i = threadIdx.x; i < NB_F; i += BLK) af += pf[i];
    for (int i = threadIdx.x; i < NB_B; i += BLK) { ag += pg[i]; ac += pc[i]; }

    __shared__ float smem[BLK / 32];
    const float tf = block_sum(af, smem);
    const float tg = block_sum(ag, smem);
    const float tc = block_sum(ac, smem);

    if (threadIdx.x == 0) {
        const float num_fg = fmaxf(tc, 1.0f);
        const float ll = tf / num_fg;   // loss_labels
        const float lb = tg / num_fg;   // loss_bboxes
        out[0] = ll;
        out[1] = lb;
        out[2] = ll + lb;               // LOSS_CLS_W=LOSS_REG_W=1
    }
}

extern "C" void kernel_launch(void* const* d_in, const int* in_sizes, int n_in,
                              void* d_out, int out_size, void* d_ws, size_t ws_size,
                              hipStream_t stream) {
    const float4*        pcls = (const float4*)d_in[0];        // (B,M,80) f32
    const float4*        pbox = (const float4*)d_in[1];        // (B,M,4)  f32
    const int*           tcls = (const int*)d_in[2];           // (B,M)    i32
    const float4*        tbox = (const float4*)d_in[3];        // (B,M,4)  f32
    const unsigned char* mask = (const unsigned char*)d_in[4]; // (B,M)    bool (1B)
    float* out = (float*)d_out;                                // 3 floats
    float* ws  = (float*)d_ws;

    float* pf = ws;                 // NB_F focal partials
    float* pg = ws + NB_F;          // NB_B giou partials
    float* pc = ws + NB_F + NB_B;   // NB_B fg-count partials

    const int n4 = in_sizes[0] / 4;   // float4 count of pred_cls
    const int na = in_sizes[2];       // number of anchors (B*M)

    focal_kernel<<<NB_F, BLK, 0, stream>>>(pcls, tcls, mask, pf, n4);
    box_kernel<<<NB_B, BLK, 0, stream>>>(pbox, tbox, tcls, pg, pc, na);
    finalize_kernel<<<1, BLK, 0, stream>>>(pf, pg, pc, out);
}